// Wkv7_26740466385134
// MI455X (gfx1250) — compile-verified
//
#include <hip/hip_runtime.h>

typedef float v8f __attribute__((ext_vector_type(8)));
typedef float v4f __attribute__((ext_vector_type(4)));
typedef float v2f __attribute__((ext_vector_type(2)));

namespace {
constexpr int T = 4096;
constexpr int H = 32;
constexpr int N = 64;
constexpr int HN = H * N;
}

// One block per head (32 blocks), 128 threads = 4 waves (wave32).
// Wave `wi` owns state rows [wi*16, wi*16+16) as 4 WMMA f32 16x16 accumulator
// tiles (v8f each). Element mapping (ISA C/D layout): VGPR vv, lane l holds
// (m = wi*16 + vv + 8*(l>>4), j = bj*16 + (l&15)).
__global__ __launch_bounds__(128) void wkv7_kernel(
    const float* __restrict__ g_r, const float* __restrict__ g_w,
    const float* __restrict__ g_k, const float* __restrict__ g_v,
    const float* __restrict__ g_a, const float* __restrict__ g_b,
    const float* __restrict__ g_s0, float* __restrict__ g_x,
    float* __restrict__ g_sout) {
  // per-wave padded scratch: rows 0..15 = sa partials, rows 16..31 = x partials
  // (stride 20 floats = 80B: 16B-aligned rows, conflict-free column stores)
  __shared__ float red[4][32][20];

  const int head = blockIdx.x;
  const int wave = threadIdx.x >> 5;  // row-block bi
  const int lane = threadIdx.x & 31;
  const int n16 = lane & 15;
  const int h16 = lane >> 4;

  const float* pr = g_r + head * N;
  const float* pw = g_w + head * N;
  const float* pk = g_k + head * N;
  const float* pv = g_v + head * N;
  const float* pa = g_a + head * N;
  const float* pb = g_b + head * N;

  // ---- load initial state directly into WMMA accumulator layout ----
  v8f acc[4];
#pragma unroll
  for (int bj = 0; bj < 4; ++bj) {
#pragma unroll
    for (int vv = 0; vv < 8; ++vv) {
      const int m = wave * 16 + vv + 8 * h16;
      acc[bj][vv] = g_s0[head * N * N + m * N + bj * 16 + n16];
    }
  }

  // One time step. Computes with `c*` operands, preloads step t+1 into `n*`.
  // All hot-loop indices are 32-bit so uniform parts stay in SGPRs.
  auto step = [&](int t, float (&ca)[4], float (&cw)[4], float (&cb)[4],
                  float (&ck)[4], float (&cr)[4], float& cv, float (&na)[4],
                  float (&nw)[4], float (&nb)[4], float (&nk)[4],
                  float (&nr)[4], float& nv) {
    // ---- register prefetch of next step's operands ----
    {
      const int o = ((t + 1 < T) ? t + 1 : T - 1) * HN;
#pragma unroll
      for (int bj = 0; bj < 4; ++bj) {
        const int j = o + bj * 16 + n16;
        na[bj] = pa[j];
        nw[bj] = pw[j];
        nb[bj] = pb[j];
        nk[bj] = pk[j];
        nr[bj] = pr[j];
      }
      nv = pv[o + wave * 16 + n16];
    }
    // ---- cache prefetch ~8 steps ahead; lanes fan out to cover 256B ----
    {
      const int o = ((t + 8 < T) ? t + 8 : T - 1) * HN + lane * 2;
      __builtin_prefetch(&pa[o], 0, 1);
      __builtin_prefetch(&pw[o], 0, 1);
      __builtin_prefetch(&pb[o], 0, 1);
      __builtin_prefetch(&pk[o], 0, 1);
      __builtin_prefetch(&pr[o], 0, 1);
      __builtin_prefetch(&pv[o], 0, 1);
    }

    // ---- per-lane dot partials from OLD state ----
    // sa = S_old@a ; xr = (S_old*diag(w))@r ; plus scalar dots b.r and k.r
    float wr[4];
#pragma unroll
    for (int bj = 0; bj < 4; ++bj) wr[bj] = cw[bj] * cr[bj];
    float brp = cb[0] * cr[0];
    float krp = ck[0] * cr[0];
#pragma unroll
    for (int bj = 1; bj < 4; ++bj) {
      brp = fmaf(cb[bj], cr[bj], brp);
      krp = fmaf(ck[bj], cr[bj], krp);
    }
    float sp[8], xp[8];
#pragma unroll
    for (int vv = 0; vv < 8; ++vv) {
      float s = acc[0][vv] * ca[0];
      float xx = acc[0][vv] * wr[0];
#pragma unroll
      for (int bj = 1; bj < 4; ++bj) {
        s = fmaf(acc[bj][vv], ca[bj], s);
        xx = fmaf(acc[bj][vv], wr[bj], xx);
      }
      sp[vv] = s;
      xp[vv] = xx;
    }
    // single LDS transpose-reduce phase (in-wave DS ordering, no barrier)
#pragma unroll
    for (int vv = 0; vv < 8; ++vv) {
      red[wave][vv + 8 * h16][n16] = sp[vv];
      red[wave][16 + vv + 8 * h16][n16] = xp[vv];
    }

    // butterfly-reduce b.r and k.r across each 16-lane half (overlaps LDS)
#pragma unroll
    for (int m = 1; m < 16; m <<= 1) {
      brp += __shfl_xor(brp, m, 32);
      krp += __shfl_xor(krp, m, 32);
    }
    // column decay on old state (also overlaps LDS latency)
#pragma unroll
    for (int bj = 0; bj < 4; ++bj) {
#pragma unroll
      for (int vv = 0; vv < 8; ++vv) acc[bj][vv] *= cw[bj];
    }

    // ---- row sums: sa[wave*16+n16] and xr[wave*16+n16] ----
    float sa_l, xr_l;
    {
      const v4f* rs = (const v4f*)&red[wave][n16][0];
      const v4f* rx = (const v4f*)&red[wave][16 + n16][0];
      v4f s0 = rs[0] + rs[1] + rs[2] + rs[3];
      v4f x0 = rx[0] + rx[1] + rx[2] + rx[3];
      sa_l = (s0.x + s0.y) + (s0.z + s0.w);
      xr_l = (x0.x + x0.y) + (x0.z + x0.w);
    }

    // x_t[m] = (S_old*w)@r + sa[m]*(b.r) + v[m]*(k.r)   (exact expansion)
    // Both lane-halves hold identical x_l for the same address: the
    // duplicate same-value store is deterministic and avoids EXEC juggling.
    const float x_l = fmaf(sa_l, brp, fmaf(cv, krp, xr_l));
    g_x[t * HN + head * N + wave * 16 + n16] = x_l;

    // ---- rank-2 update via WMMA: S = S*diag(w) + sa*b^T + v*k^T ----
    // A (16x4): col0 = sa, col3 = v. A layout: VGPR0={K0|K2}, VGPR1={K1|K3}.
    v2f A;
    A.x = (h16 == 0) ? sa_l : 0.0f;  // A[m,0]=sa[m], A[m,2]=0
    A.y = (h16 == 0) ? 0.0f : cv;    // A[m,1]=0,     A[m,3]=v[m]
#pragma unroll
    for (int bj = 0; bj < 4; ++bj) {
      // B (4x16): row0 = b, row3 = k. Slots (VGPR0,h0)->K0, (VGPR1,h1)->K3
      // under either plausible B layout.
      v2f B;
      B.x = (h16 == 0) ? cb[bj] : 0.0f;
      B.y = (h16 == 0) ? 0.0f : ck[bj];
      acc[bj] = __builtin_amdgcn_wmma_f32_16x16x4_f32(
          false, A, false, B, (short)0, acc[bj], false, false);
    }
  };

  // ping-pong operand buffers; loop unrolled x2 so no register rotation
  float a0[4], w0[4], b0[4], k0[4], r0[4], v0;
  float a1[4], w1[4], b1[4], k1[4], r1[4], v1;
  {
#pragma unroll
    for (int bj = 0; bj < 4; ++bj) {
      const int j = bj * 16 + n16;
      a0[bj] = pa[j];
      w0[bj] = pw[j];
      b0[bj] = pb[j];
      k0[bj] = pk[j];
      r0[bj] = pr[j];
    }
    v0 = pv[wave * 16 + n16];
  }

  for (int t = 0; t < T; t += 2) {
    step(t, a0, w0, b0, k0, r0, v0, a1, w1, b1, k1, r1, v1);
    step(t + 1, a1, w1, b1, k1, r1, v1, a0, w0, b0, k0, r0, v0);
  }

  // ---- write final state ----
#pragma unroll
  for (int bj = 0; bj < 4; ++bj) {
#pragma unroll
    for (int vv = 0; vv < 8; ++vv) {
      const int m = wave * 16 + vv + 8 * h16;
      g_sout[head * N * N + m * N + bj * 16 + n16] = acc[bj][vv];
    }
  }
}

extern "C" void kernel_launch(void* const* d_in, const int* in_sizes, int n_in,
                              void* d_out, int out_size, void* d_ws,
                              size_t ws_size, hipStream_t stream) {
  // setup_inputs order: 0=seq_length(int), 1=r, 2=w, 3=k, 4=v, 5=a, 6=b, 7=state2
  const float* r = (const float*)d_in[1];
  const float* w = (const float*)d_in[2];
  const float* k = (const float*)d_in[3];
  const float* v = (const float*)d_in[4];
  const float* a = (const float*)d_in[5];
  const float* b = (const float*)d_in[6];
  const float* s0 = (const float*)d_in[7];

  float* x = (float*)d_out;           // [T, H, 1, N]
  float* sout = x + (long)T * H * N;  // [H, N, N] appended

  wkv7_kernel<<<dim3(H), dim3(128), 0, stream>>>(r, w, k, v, a, b, s0, x, sout);
}